// GGNN_70136815944018
// MI455X (gfx1250) — compile-verified
//
#include <hip/hip_runtime.h>
#include <hip/hip_bf16.h>

#define N_NODES   100000
#define N_EDGES   1600000
#define C         128
#define LAYERS    3
#define NUM_CLASS 10
#define NUM_GRAPHS 64

typedef __attribute__((ext_vector_type(2))) float v2f;
typedef __attribute__((ext_vector_type(8))) float v8f;

__device__ __forceinline__ v8f wmma_f32(v2f a, v2f b, v8f c) {
    // D = A(16x4) * B(4x16) + C, all f32, exact FMA math (matches f32 reference)
    return __builtin_amdgcn_wmma_f32_16x16x4_f32(
        /*neg_a=*/false, a, /*neg_b=*/false, b,
        /*c_mod=*/(short)0, c, /*reuse_a=*/false, /*reuse_b=*/false);
}

__device__ __forceinline__ float sigmoidf_(float x) {
    return 1.0f / (1.0f + __expf(-x));
}

// ---------------------------------------------------------------------------
// Pack W[l] (row-major [k][o]) into K-pair-interleaved layout so the gemm
// B-fragment {W[k][o], W[k+1][o]} becomes one contiguous, aligned b64 load:
//   Wp[l][(p*C + o)*2 + {0,1}] = { W[l][2p][o], W[l][2p+1][o] },  p = k/2
// ---------------------------------------------------------------------------
__global__ void pack_weights_kernel(const float* __restrict__ W,
                                    float* __restrict__ Wp)
{
    const int idx = blockIdx.x * 256 + threadIdx.x;      // over L*(C/2)*C
    if (idx >= LAYERS * (C / 2) * C) return;
    const int l = idx / ((C / 2) * C);
    const int r = idx % ((C / 2) * C);
    const int p = r / C;
    const int o = r % C;
    const float* w = W + (size_t)l * C * C;
    float* dst = Wp + (size_t)l * C * C + ((size_t)p * C + o) * 2;
    dst[0] = w[(2 * p)     * C + o];
    dst[1] = w[(2 * p + 1) * C + o];
}

// ---------------------------------------------------------------------------
// m = h @ W   (N x 128) @ (128 x 128), W pre-packed (see above)
// block: 256 threads = 8 waves; block handles 16 rows, wave w -> col tile w
// ---------------------------------------------------------------------------
__global__ __launch_bounds__(256) void gemm_m_kernel(
    const float* __restrict__ h, const float* __restrict__ Wp,
    float* __restrict__ m)
{
    __shared__ float hs[16 * C];            // 8 KB row block of h
    const int row0 = blockIdx.x * 16;

    for (int i = threadIdx.x; i < 16 * C / 4; i += 256)
        ((float4*)hs)[i] = ((const float4*)(h + (size_t)row0 * C))[i];
    __syncthreads();

    const int wave = threadIdx.x >> 5;
    const int lane = threadIdx.x & 31;
    const int n    = lane & 15;             // A-row / B-col within tile
    const int kh   = lane >> 4;             // K-half select (0/1)
    const int k0   = kh * 2;
    const int col0 = wave * 16;

    // base: pair p = (k>>1) + kh, element offset (p*C + col0+n)*2
    const float* wp = Wp + ((size_t)kh * C + col0 + n) * 2;

    v8f acc = {};
    #pragma unroll 4
    for (int k = 0; k < C; k += 4) {
        v2f a = *(const v2f*)&hs[n * C + k + k0];
        v2f b = *(const v2f*)&wp[(size_t)(k >> 1) * C * 2];
        acc = wmma_f32(a, b, acc);
    }

    const int rbase = 8 * kh;
    #pragma unroll
    for (int v = 0; v < 8; ++v)
        m[(size_t)(row0 + rbase + v) * C + col0 + n] = acc[v];
}

// ---------------------------------------------------------------------------
// agg[dst] += m[src] : one wave per edge, lane handles 4 floats
// ---------------------------------------------------------------------------
__global__ __launch_bounds__(256) void scatter_add_kernel(
    const float* __restrict__ m,
    const int* __restrict__ src, const int* __restrict__ dst,
    float* __restrict__ agg)
{
    const int e    = blockIdx.x * 8 + (threadIdx.x >> 5);
    const int lane = threadIdx.x & 31;
    const int s = src[e];
    const int d = dst[e];
    const float4 v = *(const float4*)(m + (size_t)s * C + lane * 4);
    float* out = agg + (size_t)d * C + lane * 4;
    atomicAdd(out + 0, v.x);
    atomicAdd(out + 1, v.y);
    atomicAdd(out + 2, v.z);
    atomicAdd(out + 3, v.w);
}

// ---------------------------------------------------------------------------
// Fused GRU cell, in-place on h (row-local dataflow):
//   gi = agg @ w_ih^T, gh = h @ w_hh^T  (both 16 x 384 per block, in regs)
//   r = sig(ir+hr), z = sig(iz+hz), n = tanh(in + r*hn), h = (1-z)*n + z*h
// block: 256 threads = 8 waves; wave w owns within-gate col tile w (16 cols)
// ---------------------------------------------------------------------------
__global__ __launch_bounds__(256) void gru_fused_kernel(
    float* __restrict__ h, const float* __restrict__ agg,
    const float* __restrict__ w_ih, const float* __restrict__ w_hh,
    const float* __restrict__ b_ih, const float* __restrict__ b_hh)
{
    __shared__ float as_[16 * C];
    __shared__ float hs_[16 * C];
    const int row0 = blockIdx.x * 16;

    for (int i = threadIdx.x; i < 16 * C / 4; i += 256) {
        ((float4*)as_)[i] = ((const float4*)(agg + (size_t)row0 * C))[i];
        ((float4*)hs_)[i] = ((const float4*)(h   + (size_t)row0 * C))[i];
    }
    __syncthreads();

    const int wave = threadIdx.x >> 5;
    const int lane = threadIdx.x & 31;
    const int n    = lane & 15;
    const int k0   = (lane >> 4) * 2;
    const int cn   = wave * 16 + n;         // within-gate output column

    __builtin_prefetch(w_ih + (size_t)cn * C, 0, 3);
    __builtin_prefetch(w_hh + (size_t)cn * C, 0, 3);

    v8f ir = {}, iz = {}, in_ = {}, hr = {}, hz = {}, hn = {};

    // gi[i,o] = sum_k agg[i,k] * w_ih[o,k] -> B fragments are contiguous b64
    const float* wr_i = w_ih + (size_t)(0 * C + cn) * C;
    const float* wz_i = w_ih + (size_t)(1 * C + cn) * C;
    const float* wn_i = w_ih + (size_t)(2 * C + cn) * C;
    const float* wr_h = w_hh + (size_t)(0 * C + cn) * C;
    const float* wz_h = w_hh + (size_t)(1 * C + cn) * C;
    const float* wn_h = w_hh + (size_t)(2 * C + cn) * C;

    #pragma unroll 2
    for (int k = 0; k < C; k += 4) {
        v2f aa = *(const v2f*)&as_[n * C + k + k0];
        v2f ah = *(const v2f*)&hs_[n * C + k + k0];
        v2f br_i = *(const v2f*)&wr_i[k + k0];
        v2f bz_i = *(const v2f*)&wz_i[k + k0];
        v2f bn_i = *(const v2f*)&wn_i[k + k0];
        v2f br_h = *(const v2f*)&wr_h[k + k0];
        v2f bz_h = *(const v2f*)&wz_h[k + k0];
        v2f bn_h = *(const v2f*)&wn_h[k + k0];
        ir  = wmma_f32(aa, br_i, ir);
        iz  = wmma_f32(aa, bz_i, iz);
        in_ = wmma_f32(aa, bn_i, in_);
        hr  = wmma_f32(ah, br_h, hr);
        hz  = wmma_f32(ah, bz_h, hz);
        hn  = wmma_f32(ah, bn_h, hn);
    }

    const float bir = b_ih[0 * C + cn], bhr = b_hh[0 * C + cn];
    const float biz = b_ih[1 * C + cn], bhz = b_hh[1 * C + cn];
    const float bin = b_ih[2 * C + cn], bhn = b_hh[2 * C + cn];

    const int rbase = 8 * (lane >> 4);
    #pragma unroll
    for (int v = 0; v < 8; ++v) {
        const int row = rbase + v;
        const float hv = hs_[row * C + cn];
        const float r  = sigmoidf_((ir[v] + bir) + (hr[v] + bhr));
        const float z  = sigmoidf_((iz[v] + biz) + (hz[v] + bhz));
        const float nn = tanhf((in_[v] + bin) + r * (hn[v] + bhn));
        h[(size_t)(row0 + row) * C + cn] = (1.0f - z) * nn + z * hv;
    }
}

// ---------------------------------------------------------------------------
// L2-normalize + ReLU + segment max/sum/count pooling (one node per block)
// ---------------------------------------------------------------------------
__global__ __launch_bounds__(128) void norm_pool_kernel(
    const float* __restrict__ h, const int* __restrict__ batch,
    float* __restrict__ pmax, float* __restrict__ psum, float* __restrict__ pcnt)
{
    __shared__ float red[C];
    const int i = blockIdx.x;
    const int c = threadIdx.x;
    const float v = h[(size_t)i * C + c];
    red[c] = v * v;
    __syncthreads();
    #pragma unroll
    for (int s = 64; s > 0; s >>= 1) {
        if (c < s) red[c] += red[c + s];
        __syncthreads();
    }
    const float norm = fmaxf(sqrtf(red[0]), 1e-12f);
    const float o = fmaxf(v, 0.0f) / norm;

    const int g = batch[i];
    // o >= 0 and pmax zero-initialized -> uint compare == float compare
    atomicMax((unsigned int*)&pmax[g * C + c], __float_as_uint(o));
    atomicAdd(&psum[g * C + c], o);
    if (c == 0) atomicAdd(&pcnt[g], 1.0f);
}

// ---------------------------------------------------------------------------
// out[g][cls] = [pmax(g) , psum(g)/cnt(g)] . lin_w[cls] + lin_b[cls]
// ---------------------------------------------------------------------------
__global__ void final_linear_kernel(
    const float* __restrict__ pmax, const float* __restrict__ psum,
    const float* __restrict__ pcnt,
    const float* __restrict__ lin_w, const float* __restrict__ lin_b,
    float* __restrict__ out)
{
    const int g = blockIdx.x;
    const int t = threadIdx.x;
    if (t >= NUM_CLASS) return;
    const float inv_cnt = 1.0f / fmaxf(pcnt[g], 1.0f);
    float acc = lin_b[t];
    for (int j = 0; j < C; ++j) {
        acc += pmax[g * C + j] * lin_w[t * (2 * C) + j];
        acc += (psum[g * C + j] * inv_cnt) * lin_w[t * (2 * C) + C + j];
    }
    out[g * NUM_CLASS + t] = acc;
}

__global__ void fill_zero4_kernel(float4* __restrict__ p, int n4) {
    int i = blockIdx.x * blockDim.x + threadIdx.x;
    if (i < n4) p[i] = make_float4(0.f, 0.f, 0.f, 0.f);
}

__global__ void fill_zero_kernel(float* __restrict__ p, int n) {
    int i = blockIdx.x * blockDim.x + threadIdx.x;
    if (i < n) p[i] = 0.0f;
}

// ---------------------------------------------------------------------------
extern "C" void kernel_launch(void* const* d_in, const int* in_sizes, int n_in,
                              void* d_out, int out_size, void* d_ws, size_t ws_size,
                              hipStream_t stream) {
    const float* x      = (const float*)d_in[0];
    const int*   eidx   = (const int*)  d_in[1];     // [2, E]
    const int*   batch  = (const int*)  d_in[2];
    const float* weight = (const float*)d_in[3];     // [L, C, C]
    const float* w_ih   = (const float*)d_in[4];
    const float* w_hh   = (const float*)d_in[5];
    const float* b_ih   = (const float*)d_in[6];
    const float* b_hh   = (const float*)d_in[7];
    const float* lin_w  = (const float*)d_in[8];
    const float* lin_b  = (const float*)d_in[9];
    float* out = (float*)d_out;

    const int* src = eidx;
    const int* dst = eidx + N_EDGES;

    const size_t NC = (size_t)N_NODES * C;
    float* ws    = (float*)d_ws;
    float* h     = ws;                 // N*C
    float* m     = ws + NC;            // N*C
    float* agg   = ws + 2 * NC;        // N*C
    float* pmax  = ws + 3 * NC;        // G*C
    float* psum  = pmax + NUM_GRAPHS * C;
    float* pcnt  = psum + NUM_GRAPHS * C;         // G
    float* wpack = pcnt + NUM_GRAPHS;             // L*C*C packed gemm weights

    // h = x (do not mutate inputs)
    hipMemcpyAsync(h, x, NC * sizeof(float), hipMemcpyDeviceToDevice, stream);

    // one-shot weight repack for the m-GEMM (deterministic, every launch)
    pack_weights_kernel<<<(LAYERS * (C / 2) * C + 255) / 256, 256, 0, stream>>>(
        weight, wpack);

    const int nRowBlocks  = N_NODES / 16;            // 6250, exact
    const int nEdgeBlocks = N_EDGES / 8;             // 200000, exact (wave/edge)
    const int nAgg4 = (int)(NC / 4);                 // float4 count

    for (int l = 0; l < LAYERS; ++l) {
        gemm_m_kernel<<<nRowBlocks, 256, 0, stream>>>(h, wpack + (size_t)l * C * C, m);
        fill_zero4_kernel<<<(nAgg4 + 255) / 256, 256, 0, stream>>>((float4*)agg, nAgg4);
        scatter_add_kernel<<<nEdgeBlocks, 256, 0, stream>>>(m, src, dst, agg);
        gru_fused_kernel<<<nRowBlocks, 256, 0, stream>>>(h, agg, w_ih, w_hh, b_ih, b_hh);
    }

    const int nPool = NUM_GRAPHS * C * 2 + NUM_GRAPHS;
    fill_zero_kernel<<<(nPool + 255) / 256, 256, 0, stream>>>(pmax, nPool);
    norm_pool_kernel<<<N_NODES, 128, 0, stream>>>(h, batch, pmax, psum, pcnt);
    final_linear_kernel<<<NUM_GRAPHS, 32, 0, stream>>>(pmax, psum, pcnt, lin_w, lin_b, out);
}